// QSVTLayer_19258633355995
// MI455X (gfx1250) — compile-verified
//
#include <hip/hip_runtime.h>
#include <stdint.h>
#include <math.h>

// ---------------------------------------------------------------------------
// QSVT elementwise kernel for gfx1250 (MI455X).
//
// Math: the whole 8-phase QSVT chain collapses exactly to
//     out(x) = E(u) + cos(x)*Cc(u) + sin(x)*sqrt(1-u)*x*g(u),  u = x^2
// with E,Cc (8 coeffs) and g (7 coeffs) depending only on phi.
//  - setup kernel (1 thread, fp64, rolled loops) builds 23 coeffs into d_ws
//  - main kernel: ~35 VALU + 3 TRANS per element, three independent Horner
//    chains (VOPD dual-issue friendly), coefficients via uniform loads
//  - input staged via Tensor Data Mover (tensor_load_to_lds), double
//    buffered in LDS, synced with s_wait_tensorcnt
//  - theta and the last PCPhase are magnitude-invariant -> dropped exactly
// ---------------------------------------------------------------------------

#define TOTAL_ELEMS (4096u * 4096u)
#define WGS   256
#define TILE  4096u                       // floats per TDM tile = 16 KB
#define GRID  1024u
#define NT    (TOTAL_ELEMS / (TILE * GRID))   // = 4 tiles per workgroup

typedef unsigned int uint32x4 __attribute__((ext_vector_type(4)));
typedef int          int32x4  __attribute__((ext_vector_type(4)));
typedef int          int32x8  __attribute__((ext_vector_type(8)));

// ---------------------------------------------------------------------------
// TDM: build a D# (CDNA5 ISA) for a 1-D row of `n` fp32 and fire
// TENSOR_LOAD_TO_LDS. n must be <= 65535 (tile_dim0 is 16-bit).
// ---------------------------------------------------------------------------
__device__ __forceinline__ void tdm_load_tile_f32(const float* gsrc,
                                                  uint32_t lds_byte_off,
                                                  uint32_t n) {
  uint64_t ga = (uint64_t)(uintptr_t)gsrc;

  uint32x4 g0;
  g0.x = 1u;                                            // count=1, user mode
  g0.y = lds_byte_off;                                  // lds_addr[31:0]
  g0.z = (uint32_t)ga;                                  // global_addr[31:0]
  g0.w = ((uint32_t)(ga >> 32) & 0x01FFFFFFu)           // global_addr[56:32]
       | (2u << 30);                                    // type = 2 ("image")

  int32x8 g1;
  g1.s0 = (int)(2u << 16);                              // data_size = 4 bytes
  g1.s1 = (int)(n << 16);                               // tensor_dim0[15:0]
  g1.s2 = (int)(((n >> 16) & 0xFFFFu) | (1u << 16));    // tensor_dim0[31:16], tensor_dim1=1
  g1.s3 = (int)(n << 16);                               // tile_dim0 = n (1-D tile)
  g1.s4 = 0;                                            // tile_dim1 = tile_dim2 = 0
  g1.s5 = (int)n;                                       // tensor_dim0_stride (unused 1-D)
  g1.s6 = 0;
  g1.s7 = 0;

  int32x4 z4 = {0, 0, 0, 0};
#if __clang_major__ >= 23
  int32x8 z8 = {0, 0, 0, 0, 0, 0, 0, 0};
  __builtin_amdgcn_tensor_load_to_lds(g0, g1, z4, z4, z8, 0);
#else
  __builtin_amdgcn_tensor_load_to_lds(g0, g1, z4, z4, 0);
#endif
}

// ---------------------------------------------------------------------------
// Setup: evolve W = (U P_6)(U P_5)...(U P_0) symbolically.
//   W00 = A(x) [odd,deg7]  W01 = s*B(x) [B even,deg6]
//   W10 = s*C(x) [C even]  W11 = D(x)  [odd,deg7]
// step (p = e^{i phi}):
//   A' = x pA + (1-x^2) p̄C ; C' = pA - x p̄C
//   B' = x pB + p̄D         ; D' = (1-x^2) pB - x p̄D
// Then F1=|A|^2-(1-x^2)|C|^2, F2=(1-x^2)|B|^2-|D|^2,
//      E=(F1+F2)/2, Cc=(F1-F2)/2 (even -> polys in u), x*g(u)=Re(AB̄)-Re(CD̄).
// coef layout: [0..7]=E, [8..15]=Cc, [16..22]=g   (23 floats)
// Runs once on one thread; loops kept rolled to keep code compact (no spills
// blow-up); runtime is negligible either way.
// ---------------------------------------------------------------------------
__global__ __launch_bounds__(32) void qsvt_setup(const float* __restrict__ phi,
                                                 float* __restrict__ coef) {
  if (threadIdx.x != 0 || blockIdx.x != 0) return;

  double Ar[8] = {}, Ai[8] = {}, Br[8] = {}, Bi[8] = {};
  double Cr[8] = {}, Ci[8] = {}, Dr[8] = {}, Di[8] = {};
  Ar[0] = 1.0; Dr[0] = 1.0;                 // W = I

#pragma clang loop unroll(disable)
  for (int k = 0; k < 7; ++k) {
    double pc = cos((double)phi[k]), ps = sin((double)phi[k]);
    double nAr[8] = {}, nAi[8] = {}, nBr[8] = {}, nBi[8] = {};
    double nCr[8] = {}, nCi[8] = {}, nDr[8] = {}, nDi[8] = {};
#pragma clang loop unroll(disable)
    for (int n = 0; n < 8; ++n) {
      double pAr = pc * Ar[n] - ps * Ai[n], pAi = pc * Ai[n] + ps * Ar[n];
      double qCr = pc * Cr[n] + ps * Ci[n], qCi = pc * Ci[n] - ps * Cr[n];
      double pBr = pc * Br[n] - ps * Bi[n], pBi = pc * Bi[n] + ps * Br[n];
      double qDr = pc * Dr[n] + ps * Di[n], qDi = pc * Di[n] - ps * Dr[n];
      if (n + 1 < 8) { nAr[n+1] += pAr; nAi[n+1] += pAi; }      // x*pA
      nAr[n] += qCr; nAi[n] += qCi;                              // +p̄C
      if (n + 2 < 8) { nAr[n+2] -= qCr; nAi[n+2] -= qCi; }      // -x^2*p̄C
      nCr[n] += pAr; nCi[n] += pAi;                              // pA
      if (n + 1 < 8) { nCr[n+1] -= qCr; nCi[n+1] -= qCi; }      // -x*p̄C
      if (n + 1 < 8) { nBr[n+1] += pBr; nBi[n+1] += pBi; }      // x*pB
      nBr[n] += qDr; nBi[n] += qDi;                              // +p̄D
      nDr[n] += pBr; nDi[n] += pBi;                              // pB
      if (n + 2 < 8) { nDr[n+2] -= pBr; nDi[n+2] -= pBi; }      // -x^2*pB
      if (n + 1 < 8) { nDr[n+1] -= qDr; nDi[n+1] -= qDi; }      // -x*p̄D
    }
#pragma clang loop unroll(disable)
    for (int n = 0; n < 8; ++n) {
      Ar[n] = nAr[n]; Ai[n] = nAi[n]; Br[n] = nBr[n]; Bi[n] = nBi[n];
      Cr[n] = nCr[n]; Ci[n] = nCi[n]; Dr[n] = nDr[n]; Di[n] = nDi[n];
    }
  }

  double F1[16] = {}, F2[16] = {}, Gx[16] = {}, CC[16] = {}, BB[16] = {};
#pragma clang loop unroll(disable)
  for (int i = 0; i < 8; ++i)
#pragma clang loop unroll(disable)
    for (int j = 0; j < 8; ++j) {
      F1[i+j] += Ar[i] * Ar[j] + Ai[i] * Ai[j];                  // |A|^2
      CC[i+j] += Cr[i] * Cr[j] + Ci[i] * Ci[j];                  // |C|^2
      BB[i+j] += Br[i] * Br[j] + Bi[i] * Bi[j];                  // |B|^2
      F2[i+j] -= Dr[i] * Dr[j] + Di[i] * Di[j];                  // -|D|^2
      Gx[i+j] += Ar[i] * Br[j] + Ai[i] * Bi[j];                  // +Re(A B̄)
      Gx[i+j] -= Cr[i] * Dr[j] + Ci[i] * Di[j];                  // -Re(C D̄)
    }
#pragma clang loop unroll(disable)
  for (int n = 0; n < 14; ++n) {                                 // *(1-x^2)
    F1[n] -= CC[n]; F1[n+2] += CC[n];
    F2[n] += BB[n]; F2[n+2] -= BB[n];
  }
  F1[14] -= CC[14]; F2[14] += BB[14];                            // (zero anyway)

#pragma clang loop unroll(disable)
  for (int j = 0; j < 8; ++j) {
    coef[j]     = (float)(0.5 * (F1[2*j] + F2[2*j]));            // E
    coef[8 + j] = (float)(0.5 * (F1[2*j] - F2[2*j]));            // Cc
  }
#pragma clang loop unroll(disable)
  for (int j = 0; j < 7; ++j) coef[16 + j] = (float)Gx[2*j + 1]; // g
}

// ---------------------------------------------------------------------------
// Main kernel
// ---------------------------------------------------------------------------
__device__ __forceinline__ float qsvt_eval(float x, const float* E,
                                           const float* Cc, const float* g) {
  x = fminf(fmaxf(x, -1.0f), 1.0f);
  float u  = x * x;
  float s  = sqrtf(fmaxf(1.0f - u, 0.0f));
  float cx = __cosf(x);
  float sx = __sinf(x);
  // three independent Horner chains in u (VOPD-pairable)
  float e = E[7], c = Cc[7], gg = g[6];
#pragma unroll
  for (int j = 6; j >= 0; --j) {
    e = fmaf(e, u, E[j]);
    c = fmaf(c, u, Cc[j]);
    if (j >= 1) gg = fmaf(gg, u, g[j - 1]);
  }
  float r = fmaf(cx, c, e);
  return fmaf(sx * s * x, gg, r);
}

__global__ __launch_bounds__(WGS) void qsvt_kernel(const float* __restrict__ x,
                                                   const float* __restrict__ cf,
                                                   float* __restrict__ out) {
  __shared__ float buf[2][TILE];          // 32 KB of the 320 KB WGP LDS pool

  const size_t   base     = (size_t)blockIdx.x * (TILE * NT);
  const bool     leader   = (threadIdx.x < 32u);   // wave 0 issues TDM ops
  const uint32_t lds_off0 = (uint32_t)(uintptr_t)&buf[0][0];
  const uint32_t lds_off1 = (uint32_t)(uintptr_t)&buf[1][0];

  // Kick off tile 0 DMA first so it overlaps coefficient loading.
  if (leader) tdm_load_tile_f32(x + base, lds_off0, TILE);

  // Uniform coefficient loads -> scalar registers.
  float E[8], Cc[8], g[7];
#pragma unroll
  for (int j = 0; j < 8; ++j) { E[j] = cf[j]; Cc[j] = cf[8 + j]; }
#pragma unroll
  for (int j = 0; j < 7; ++j) g[j] = cf[16 + j];

  for (int t = 0; t < (int)NT; ++t) {
    if (leader) {
      if (t + 1 < (int)NT) {
        // prefetch next tile into the other buffer, then wait for current one
        tdm_load_tile_f32(x + base + (size_t)(t + 1) * TILE,
                          ((t + 1) & 1) ? lds_off1 : lds_off0, TILE);
        __builtin_amdgcn_s_wait_tensorcnt(1);
      } else {
        __builtin_amdgcn_s_wait_tensorcnt(0);
      }
    }
    __syncthreads();                       // tile t visible to all waves

    const float* xb = buf[t & 1];
    float* ob = out + base + (size_t)t * TILE;
#pragma unroll
    for (int i = 0; i < (int)(TILE / (WGS * 4)); ++i) {   // 4 x float4/thread
      int idx = (i * WGS + (int)threadIdx.x) * 4;
      float4 xv = *(const float4*)(xb + idx);             // ds_load_b128
      float4 r;
      r.x = qsvt_eval(xv.x, E, Cc, g);
      r.y = qsvt_eval(xv.y, E, Cc, g);
      r.z = qsvt_eval(xv.z, E, Cc, g);
      r.w = qsvt_eval(xv.w, E, Cc, g);
      *(float4*)(ob + idx) = r;                           // global_store_b128
    }
    __syncthreads();                       // buffer t&1 free for reuse
  }
}

extern "C" void kernel_launch(void* const* d_in, const int* in_sizes, int n_in,
                              void* d_out, int out_size, void* d_ws, size_t ws_size,
                              hipStream_t stream) {
  const float* x_chunk = (const float*)d_in[0];
  // d_in[1] = theta: provably does not affect <Z> (|e^{-i theta/2}| == 1).
  const float* phi     = (const float*)d_in[2];
  float* out           = (float*)d_out;
  float* coef          = (float*)d_ws;    // 23 floats of scratch
  (void)in_sizes; (void)n_in; (void)out_size; (void)ws_size;

  qsvt_setup<<<dim3(1), dim3(1), 0, stream>>>(phi, coef);
  qsvt_kernel<<<dim3(GRID), dim3(WGS), 0, stream>>>(x_chunk, coef, out);
}